// TokenFuser_28561532518696
// MI455X (gfx1250) — compile-verified
//
#include <hip/hip_runtime.h>

typedef __attribute__((ext_vector_type(2))) float v2f;
typedef __attribute__((ext_vector_type(8))) float v8f;

#define B_  64
#define N_  8
#define H_  768
#define FS_ 32
#define S_  1024

// ---------------------------------------------------------------------------
// Kernel 1: t = LN2( Linear_n( LN1(token) ) )   [B,N,H] -> workspace
// One block per batch; 8 waves = 8 token rows (wave32 => warp-per-row LN).
// ---------------------------------------------------------------------------
__global__ __launch_bounds__(256) void tf_token_kernel(
    const float* __restrict__ token,
    const float* __restrict__ ln1w, const float* __restrict__ ln1b,
    const float* __restrict__ linw, const float* __restrict__ linb,
    const float* __restrict__ ln2w, const float* __restrict__ ln2b,
    float* __restrict__ t_out)
{
  __shared__ float tA[N_][H_];
  __shared__ float tB[N_][H_];
  const int b    = blockIdx.x;
  const int tid  = threadIdx.x;
  const int wave = tid >> 5;
  const int lane = tid & 31;

  // ---- load row + LN1 (one wave per token row) ----
  {
    const float* row = token + ((size_t)b * N_ + wave) * H_;
    float s = 0.f, sq = 0.f;
#pragma unroll
    for (int i = 0; i < H_ / 32; ++i) {
      float v = row[lane + 32 * i];
      tA[wave][lane + 32 * i] = v;
      s += v; sq += v * v;
    }
#pragma unroll
    for (int m = 16; m; m >>= 1) {
      s  += __shfl_xor(s,  m, 32);
      sq += __shfl_xor(sq, m, 32);
    }
    const float mu   = s * (1.f / H_);
    const float var  = sq * (1.f / H_) - mu * mu;
    const float rstd = rsqrtf(var + 1e-5f);
#pragma unroll
    for (int i = 0; i < H_ / 32; ++i) {
      int h = lane + 32 * i;
      tA[wave][h] = (tA[wave][h] - mu) * rstd * ln1w[h] + ln1b[h];
    }
  }
  __syncthreads();

  // ---- mix over token dim: tB[m,h] = lin_b[m] + sum_n lin_w[m,n]*tA[n,h] ----
  for (int idx = tid; idx < N_ * H_; idx += 256) {
    const int m = idx / H_;
    const int h = idx - m * H_;
    float acc = linb[m];
#pragma unroll
    for (int n = 0; n < N_; ++n) acc = fmaf(linw[m * N_ + n], tA[n][h], acc);
    tB[m][h] = acc;
  }
  __syncthreads();

  // ---- LN2 per row, write to workspace ----
  {
    float s = 0.f, sq = 0.f;
#pragma unroll
    for (int i = 0; i < H_ / 32; ++i) {
      float v = tB[wave][lane + 32 * i];
      s += v; sq += v * v;
    }
#pragma unroll
    for (int m = 16; m; m >>= 1) {
      s  += __shfl_xor(s,  m, 32);
      sq += __shfl_xor(sq, m, 32);
    }
    const float mu   = s * (1.f / H_);
    const float rstd = rsqrtf(sq * (1.f / H_) - mu * mu + 1e-5f);
    float* orow = t_out + ((size_t)b * N_ + wave) * H_;
#pragma unroll
    for (int i = 0; i < H_ / 32; ++i) {
      int h = lane + 32 * i;
      orow[h] = (tB[wave][h] - mu) * rstd * ln2w[h] + ln2b[h];
    }
  }
}

// ---------------------------------------------------------------------------
// Kernel 2: mix[b,n,x*32+y] = sigmoid( sum_h conv_w[n,h] * o[b,h,x,y] )
// o = LN over y (32 elems, eps = 32.0 per reference quirk) of ori^T.
// One block per (b,x): stage the 32x768 slab in LDS (row pitch 769 =>
// conflict-free y-strided reads), per-h stats over y, then thread (n,y)
// does the 768-deep dot. Reads ori exactly once from HBM.
// ---------------------------------------------------------------------------
#define SLAB_P 769  // 769 % 64 == 1 -> lanes with distinct y hit distinct banks

__global__ __launch_bounds__(256) void tf_mix_kernel(
    const float* __restrict__ ori,
    const float* __restrict__ lnow, const float* __restrict__ lnob,
    const float* __restrict__ convw,
    float* __restrict__ mix_out)
{
  __shared__ float slab[FS_][SLAB_P];  // [y][h], ~96 KB
  __shared__ float muS[H_];
  __shared__ float rsS[H_];
  __shared__ float cwS[N_][H_];        // 24 KB

  const int b   = blockIdx.x >> 5;     // 64*32 blocks
  const int x   = blockIdx.x & 31;
  const int tid = threadIdx.x;

  for (int i = tid; i < N_ * H_; i += 256) cwS[i / H_][i % H_] = convw[i];

  // load slab + per-column (h) LN stats over y; coalesced 1KB rows per step
  const float* base = ori + ((size_t)b * S_ + (size_t)x * FS_) * H_;
#pragma unroll
  for (int c = 0; c < H_ / 256; ++c) {
    const int h = tid + 256 * c;
    float s = 0.f, sq = 0.f;
    for (int y = 0; y < FS_; ++y) {
      float v = base[(size_t)y * H_ + h];
      slab[y][h] = v;
      s += v; sq += v * v;
    }
    const float mu  = s * (1.f / FS_);
    const float var = sq * (1.f / FS_) - mu * mu;
    muS[h] = mu;
    rsS[h] = rsqrtf(var + (float)FS_);   // eps = featmap_size (faithful)
  }
  __syncthreads();

  const int n  = tid >> 5;   // wave index -> conv output channel (broadcast reads)
  const int y  = tid & 31;
  const float lw = lnow[y];
  const float lb = lnob[y];
  float g = 0.f;
#pragma unroll 4
  for (int h = 0; h < H_; ++h) {
    const float o = fmaf((slab[y][h] - muS[h]) * rsS[h], lw, lb);
    g = fmaf(cwS[n][h], o, g);
  }
  const float m = 1.f / (1.f + __expf(-g));
  mix_out[((size_t)b * N_ + n) * S_ + x * FS_ + y] = m;
}

// ---------------------------------------------------------------------------
// Kernel 3: out[b,s,h] = sum_n t[b,n,h] * mix[b,n,s]
// Per-batch GEMM [S=1024, H=768] = A[s, n(K=8)] x B[n, h] via
// V_WMMA_F32_16X16X4_F32, two chained WMMAs per 16x16 tile.
// Grid: 64 batches x 8 s-blocks of 128. t[b] (8x768) + mix slice (8x128)
// staged in LDS with padded pitches for conflict-free operand gathers.
// ---------------------------------------------------------------------------
#define TP (H_ + 4)   // 772 % 64 == 4 -> half-wave rows land on distinct banks
#define MP 132        // 132 % 64 == 4

__global__ __launch_bounds__(256) void tf_out_kernel(
    const float* __restrict__ t_in,
    const float* __restrict__ mix_in,
    float* __restrict__ out)
{
  __shared__ float tS[N_][TP];
  __shared__ float mS[N_][MP];

  const int b    = blockIdx.x >> 3;
  const int sblk = blockIdx.x & 7;        // 128 s-rows per block
  const int tid  = threadIdx.x;

  for (int i = tid; i < N_ * H_; i += 256)
    tS[i / H_][i % H_] = t_in[(size_t)b * N_ * H_ + i];
  for (int i = tid; i < N_ * 128; i += 256) {
    const int n = i >> 7, s = i & 127;
    mS[n][s] = mix_in[((size_t)b * N_ + n) * S_ + sblk * 128 + s];
  }
  __syncthreads();

  const int wave = tid >> 5;
  const int lane = tid & 31;
  const int col  = lane & 15;
  const int koff = (lane >> 4) << 1;      // K-offset per ISA 32-bit A/B layout

  // 8 s-subtiles x 48 h-tiles = 384 tiles; uniform per-wave trip count so
  // EXEC stays all-ones around the WMMAs.
  for (int tt = wave; tt < 8 * 48; tt += 8) {
    const int st   = tt / 48;
    const int ht   = tt - st * 48;
    const int srow = st * 16 + col;
    const int h0   = ht * 16;

    // A (16x4): lane<16 holds K={0,1}, lane>=16 holds K={2,3} (VGPR0/VGPR1)
    v2f a0, a1, b0, b1;
    a0.x = mS[0 + koff][srow];      a0.y = mS[1 + koff][srow];
    a1.x = mS[4 + koff][srow];      a1.y = mS[5 + koff][srow];
    // B (4x16): same K striping, N = lane
    b0.x = tS[0 + koff][h0 + col];  b0.y = tS[1 + koff][h0 + col];
    b1.x = tS[4 + koff][h0 + col];  b1.y = tS[5 + koff][h0 + col];

    v8f acc = {};
    acc = __builtin_amdgcn_wmma_f32_16x16x4_f32(false, a0, false, b0,
                                                (short)0, acc, false, false);
    acc = __builtin_amdgcn_wmma_f32_16x16x4_f32(false, a1, false, b1,
                                                (short)0, acc, false, false);

    // D layout: VGPR v, lanes 0-15 -> M=v, lanes 16-31 -> M=v+8, N=lane&15
    const int s0 = sblk * 128 + st * 16 + ((lane >> 4) << 3);
    float* op = out + ((size_t)b * S_ + s0) * H_ + h0 + col;
#pragma unroll
    for (int v = 0; v < 8; ++v)
      op[(size_t)v * H_] = acc[v];
  }
}

// ---------------------------------------------------------------------------
extern "C" void kernel_launch(void* const* d_in, const int* in_sizes, int n_in,
                              void* d_out, int out_size, void* d_ws, size_t ws_size,
                              hipStream_t stream) {
  const float* token = (const float*)d_in[0];
  const float* ori   = (const float*)d_in[1];
  const float* ln1w  = (const float*)d_in[2];
  const float* ln1b  = (const float*)d_in[3];
  const float* linw  = (const float*)d_in[4];
  const float* linb  = (const float*)d_in[5];
  const float* ln2w  = (const float*)d_in[6];
  const float* ln2b  = (const float*)d_in[7];
  const float* lnow  = (const float*)d_in[8];
  const float* lnob  = (const float*)d_in[9];
  const float* convw = (const float*)d_in[10];

  float* out    = (float*)d_out;
  float* t_ws   = (float*)d_ws;                       // 64*8*768 f32
  float* mix_ws = t_ws + (size_t)B_ * N_ * H_;        // 64*8*1024 f32

  tf_token_kernel<<<B_, 256, 0, stream>>>(token, ln1w, ln1b, linw, linb,
                                          ln2w, ln2b, t_ws);
  tf_mix_kernel<<<B_ * FS_, 256, 0, stream>>>(ori, lnow, lnob, convw, mix_ws);
  tf_out_kernel<<<B_ * 8, 256, 0, stream>>>(t_ws, mix_ws, out);
}